// DirGCNConv_19439021982025
// MI455X (gfx1250) — compile-verified
//
#include <hip/hip_runtime.h>
#include <hip/hip_bf16.h>

typedef __attribute__((ext_vector_type(2))) float v2f;
typedef __attribute__((ext_vector_type(8))) float v8f;

#define N_NODES 50000
#define N_EDGES 800000
#define DIM     128

// ---------------- workspace layout (floats) ----------------
// [0)           .. N)            deg_row  -> inv_row (in place)
// [N)           .. 2N)           deg_col  -> inv_col (in place)
// [2N)          .. 2N+128N)      agg_fwd
// [2N+128N)     .. 2N+256N)      agg_rev
// [2N+256N)     .. +16384)       Wc_src   (ALPHA * W_src)
// [..)          .. +16384)       Wc_dst   ((1-ALPHA) * W_dst)
// [..)          .. +128)         bias     (ALPHA*b_src + (1-ALPHA)*b_dst)

__global__ void zero_kernel(float4* __restrict__ p, long n4) {
    long i = (long)blockIdx.x * blockDim.x + threadIdx.x;
    long stride = (long)gridDim.x * blockDim.x;
    float4 z = make_float4(0.f, 0.f, 0.f, 0.f);
    for (; i < n4; i += stride) p[i] = z;
}

__global__ void degree_kernel(const int* __restrict__ row,
                              const int* __restrict__ col,
                              float* __restrict__ deg_row,
                              float* __restrict__ deg_col) {
    int e = blockIdx.x * blockDim.x + threadIdx.x;
    if (e >= N_EDGES) return;
    atomicAdd(&deg_row[row[e]], 1.0f);
    atomicAdd(&deg_col[col[e]], 1.0f);
}

__global__ void invsqrt_kernel(float* __restrict__ deg) { // 2N elements in place
    int i = blockIdx.x * blockDim.x + threadIdx.x;
    if (i >= 2 * N_NODES) return;
    float d = deg[i];
    deg[i] = (d > 0.0f) ? rsqrtf(fmaxf(d, 1.0f)) : 0.0f;
}

__global__ void prep_kernel(const float* __restrict__ W_src,
                            const float* __restrict__ b_src,
                            const float* __restrict__ W_dst,
                            const float* __restrict__ b_dst,
                            float* __restrict__ Wc_s,
                            float* __restrict__ Wc_d,
                            float* __restrict__ bias) {
    int i = blockIdx.x * blockDim.x + threadIdx.x;
    if (i < DIM * DIM) {
        Wc_s[i] = 0.5f * W_src[i];   // ALPHA = 0.5
        Wc_d[i] = 0.5f * W_dst[i];   // 1 - ALPHA = 0.5
    }
    if (i < DIM) bias[i] = 0.5f * b_src[i] + 0.5f * b_dst[i];
}

// one wave32 per edge; each lane owns 4 consecutive feature floats
__global__ void scatter_kernel(const float* __restrict__ x,
                               const int* __restrict__ row,
                               const int* __restrict__ col,
                               const float* __restrict__ inv_row,
                               const float* __restrict__ inv_col,
                               float* __restrict__ agg_fwd,
                               float* __restrict__ agg_rev) {
    int lane = threadIdx.x & 31;
    int e = blockIdx.x * (blockDim.x >> 5) + (threadIdx.x >> 5);
    if (e >= N_EDGES) return;
    int r = row[e];
    int c = col[e];
    float w = inv_row[r] * inv_col[c];

    const float4 vc = ((const float4*)(x + (size_t)c * DIM))[lane];
    const float4 vr = ((const float4*)(x + (size_t)r * DIM))[lane];

    float* af = agg_fwd + (size_t)r * DIM + lane * 4;
    float* ar = agg_rev + (size_t)c * DIM + lane * 4;
    atomicAdd(af + 0, w * vc.x);
    atomicAdd(af + 1, w * vc.y);
    atomicAdd(af + 2, w * vc.z);
    atomicAdd(af + 3, w * vc.w);
    atomicAdd(ar + 0, w * vr.x);
    atomicAdd(ar + 1, w * vr.y);
    atomicAdd(ar + 2, w * vr.z);
    atomicAdd(ar + 3, w * vr.w);
}

// Fused GEMM: out = agg_fwd @ Wc_s^T + agg_rev @ Wc_d^T + bias
// One wave per 16x16 output tile; K=128 walked 4 at a time with
// V_WMMA_F32_16X16X4_F32 (f32 in, f32 accum).
__global__ void __launch_bounds__(256)
gemm_kernel(const float* __restrict__ agg_fwd,
            const float* __restrict__ agg_rev,
            const float* __restrict__ Wc_s,
            const float* __restrict__ Wc_d,
            const float* __restrict__ bias,
            float* __restrict__ out) {
    const int lane = threadIdx.x & 31;
    const int tile = blockIdx.x * 8 + (threadIdx.x >> 5); // 3125 blocks * 8 waves = 25000 tiles
    const int mt = tile >> 3;   // 0..3124  (M tiles: 50000/16)
    const int nt = tile & 7;    // 0..7     (N tiles: 128/16)

    const int half = lane >> 4;           // 0 or 1
    const int l15  = lane & 15;
    const int kb   = half * 2;            // K sub-offset this lane feeds

    const float* arow_f = agg_fwd + (size_t)(mt * 16 + l15) * DIM; // A: row M = lane&15
    const float* arow_r = agg_rev + (size_t)(mt * 16 + l15) * DIM;
    const float* wrow_s = Wc_s    + (size_t)(nt * 16 + l15) * DIM; // B col-major: col N = lane&15
    const float* wrow_d = Wc_d    + (size_t)(nt * 16 + l15) * DIM;

    v8f c;
    const float bv = bias[nt * 16 + l15]; // each lane's 8 accs share one N column
    #pragma unroll
    for (int j = 0; j < 8; ++j) c[j] = bv;

    #pragma unroll
    for (int k = 0; k < DIM; k += 4) {
        v2f a0 = *(const v2f*)(arow_f + k + kb);
        v2f b0 = *(const v2f*)(wrow_s + k + kb);
        c = __builtin_amdgcn_wmma_f32_16x16x4_f32(false, a0, false, b0,
                                                  (short)0, c, false, false);
        v2f a1 = *(const v2f*)(arow_r + k + kb);
        v2f b1 = *(const v2f*)(wrow_d + k + kb);
        c = __builtin_amdgcn_wmma_f32_16x16x4_f32(false, a1, false, b1,
                                                  (short)0, c, false, false);
    }

    // D layout: VGPR j -> row (j + half*8), col (lane&15)
    const int m0 = mt * 16 + half * 8;
    const int n  = nt * 16 + l15;
    #pragma unroll
    for (int j = 0; j < 8; ++j)
        out[(size_t)(m0 + j) * DIM + n] = c[j];
}

extern "C" void kernel_launch(void* const* d_in, const int* in_sizes, int n_in,
                              void* d_out, int out_size, void* d_ws, size_t ws_size,
                              hipStream_t stream) {
    const float* x      = (const float*)d_in[0];
    const float* W_src  = (const float*)d_in[1];
    const float* b_src  = (const float*)d_in[2];
    const float* W_dst  = (const float*)d_in[3];
    const float* b_dst  = (const float*)d_in[4];
    const int*   eidx   = (const int*)d_in[5];
    const int*   row    = eidx;
    const int*   col    = eidx + N_EDGES;
    float*       outp   = (float*)d_out;

    float* ws = (float*)d_ws;
    float* deg_row = ws;                                  // -> inv_row
    float* deg_col = ws + N_NODES;                        // -> inv_col
    float* agg_fwd = ws + 2 * (size_t)N_NODES;
    float* agg_rev = agg_fwd + (size_t)N_NODES * DIM;
    float* Wc_s    = agg_rev + (size_t)N_NODES * DIM;
    float* Wc_d    = Wc_s + DIM * DIM;
    float* bias    = Wc_d + DIM * DIM;

    // zero deg_row, deg_col, agg_fwd, agg_rev (contiguous: 258*N floats)
    {
        long n4 = (long)(2 + 2 * DIM) * N_NODES / 4;
        zero_kernel<<<2048, 256, 0, stream>>>((float4*)ws, n4);
    }
    degree_kernel<<<(N_EDGES + 255) / 256, 256, 0, stream>>>(row, col, deg_row, deg_col);
    invsqrt_kernel<<<(2 * N_NODES + 255) / 256, 256, 0, stream>>>(deg_row);
    prep_kernel<<<(DIM * DIM + 255) / 256, 256, 0, stream>>>(W_src, b_src, W_dst, b_dst,
                                                             Wc_s, Wc_d, bias);
    scatter_kernel<<<(N_EDGES + 7) / 8, 256, 0, stream>>>(x, row, col, deg_row, deg_col,
                                                          agg_fwd, agg_rev);
    gemm_kernel<<<(N_NODES / 16) * (DIM / 16) / 8, 256, 0, stream>>>(agg_fwd, agg_rev,
                                                                     Wc_s, Wc_d, bias, outp);
}